// BirthDeathIntervalLoss_19250043420933
// MI455X (gfx1250) — compile-verified
//
#include <hip/hip_runtime.h>

typedef __attribute__((ext_vector_type(2))) float v2f;
typedef __attribute__((ext_vector_type(8))) float v8f;

#define TL_H 512
#define TL_W 512
#define TL_N 64

// One wave32 per (set, 16-row tile). Rows r = b*4 + c, r in [0,128).
// WMMA A-layout (16x4 f32): lane L<16 holds A[M=L][K=0..1], lane L>=16 holds
// A[M=L-16][K=2..3]. With B = all-ones (4x16), D[m][n] = rowsum(m) for all n,
// accumulated over k0 = 0,4,...,60 -> full sum over N=64.
__global__ __launch_bounds__(32)
void topo_loss_gather_wmma(const float* __restrict__ pred,
                           const int4* __restrict__ iv0,
                           const int4* __restrict__ iv1,
                           float* __restrict__ partial)
{
    const int bid  = blockIdx.x;           // 0..15
    const int set  = bid >> 3;             // interval set 0/1
    const int tile = bid & 7;              // 8 tiles of 16 rows
    const int lane = threadIdx.x;          // 0..31, full wave
    const int m    = lane & 15;
    const int koff = (lane >> 4) << 1;     // 0 (lanes 0-15) or 2 (lanes 16-31)
    const int r    = tile * 16 + m;        // row == b*4 + c

    const int4* __restrict__ iv = set ? iv1 : iv0;
    const unsigned gpack = set ? 0x1201u : 0x3121u;  // GOOD_1 / GOOD_0, nibble per class
    const int G = (gpack >> ((r & 3) * 4)) & 15;

    const float* __restrict__ p = pred + (size_t)r * (size_t)(TL_H * TL_W);
    const int4*  __restrict__ q = iv + r * TL_N;     // intervals[b,c,n,:,:] as int4

    v8f acc_t = {};   // total sums
    v8f acc_g = {};   // good-masked sums
    v2f ones; ones.x = 1.0f; ones.y = 1.0f;

#pragma unroll
    for (int k0 = 0; k0 < TL_N; k0 += 4) {
        const int n0 = k0 + koff;
        const int4 ia = q[n0];
        const int4 ib = q[n0 + 1];
        const float b0 = p[ia.x * TL_W + ia.y];  // birth: pred[.., i00, i01]
        const float d0 = p[ia.z * TL_W + ia.w];  // death: pred[.., i10, i11]
        const float b1 = p[ib.x * TL_W + ib.y];
        const float d1 = p[ib.z * TL_W + ib.w];
        float e0 = b0 - d0; e0 *= e0;
        float e1 = b1 - d1; e1 *= e1;
        v2f at; at.x = e0; at.y = e1;
        v2f ag; ag.x = (n0 < G) ? e0 : 0.0f;
        ag.y = ((n0 + 1) < G) ? e1 : 0.0f;
        acc_t = __builtin_amdgcn_wmma_f32_16x16x4_f32(
            false, at, false, ones, (short)0, acc_t, false, false);
        acc_g = __builtin_amdgcn_wmma_f32_16x16x4_f32(
            false, ag, false, ones, (short)0, acc_g, false, false);
    }

    // D layout: VGPR v, lanes 0-15 -> M=v; lanes 16-31 -> M=v+8.
    // Every column holds the row sum, so lanes 0 and 16 have all 16 rows.
    if ((lane & 15) == 0) {
        const int half = lane >> 4;
#pragma unroll
        for (int v = 0; v < 8; ++v) {
            const int row = tile * 16 + half * 8 + v;
            const int Gr  = (gpack >> ((row & 3) * 4)) & 15;
            const float gs = acc_g[v];
            const float ts = acc_t[v];
            const float good_mean = gs / (float)(Gr > 0 ? Gr : 1);
            const float bad_mean  = (ts - gs) / (float)(TL_N - Gr);  // n_bad >= 61 > 0
            float per = 0.5f * bad_mean;                  // (1-beta) * bad_mean
            if (Gr > 0) per += 0.5f * (1.0f - good_mean); // beta * (1 - good_mean)
            // weight: alpha/(1-alpha)=0.5 per set, then /C=0.25 -> 0.125
            partial[set * 128 + row] = 0.125f * per;
        }
    }
}

// Deterministic fixed-order sum of the 256 partials (no float atomics).
__global__ void topo_loss_finalize(const float* __restrict__ partial,
                                   float* __restrict__ out)
{
    if (threadIdx.x == 0 && blockIdx.x == 0) {
        float s = 0.0f;
        for (int i = 0; i < 256; ++i) s += partial[i];
        out[0] = s;
    }
}

extern "C" void kernel_launch(void* const* d_in, const int* in_sizes, int n_in,
                              void* d_out, int out_size, void* d_ws, size_t ws_size,
                              hipStream_t stream)
{
    const float* pred = (const float*)d_in[0];
    const int4*  iv0  = (const int4*)d_in[1];   // [32,4,64,2,2] i32 -> int4 per interval
    const int4*  iv1  = (const int4*)d_in[2];
    float* partial = (float*)d_ws;              // 256 floats of scratch
    float* out     = (float*)d_out;

    topo_loss_gather_wmma<<<16, 32, 0, stream>>>(pred, iv0, iv1, partial);
    topo_loss_finalize<<<1, 32, 0, stream>>>(partial, out);
}